// YoloLayer_69114613728924
// MI455X (gfx1250) — compile-verified
//
#include <hip/hip_runtime.h>
#include <math.h>

// Problem geometry (fixed by the reference's setup_inputs)
#define NB 16
#define NA 5
#define NC 20
#define NH 64
#define NW 64
#define MAXGT 50
#define HW (NH * NW)              // 4096
#define CELLS_PER_B (NA * HW)     // 20480
#define TOTAL (NB * CELLS_PER_B)  // 327680
#define NBLOCKS (TOTAL / 256)     // 1280
#define PREP_STRIDE 12            // floats per prep entry
#define SIL_THRESH_C 0.6f
#define OBJECT_SCALE_C 5.0f

typedef __attribute__((ext_vector_type(2))) float v2f;
typedef __attribute__((ext_vector_type(8))) float v8f;

__device__ __forceinline__ float sigm(float v) { return 1.0f / (1.0f + __expf(-v)); }

// ---------------------------------------------------------------------------
// Kernel A: per-(batch, gt) prep. Computes best anchor, scatter cell, targets.
// prep entry layout (12 f32): gx gy gw gh | tx ty tw th | flat(-1 invalid) tcls pad pad
// ---------------------------------------------------------------------------
__global__ void yolo_prep(const float* __restrict__ target,
                          const float* __restrict__ anchors,
                          float* __restrict__ prep) {
  int idx = blockIdx.x * blockDim.x + threadIdx.x;
  if (idx >= NB * MAXGT) return;
  int b = idx / MAXGT;
  int g = idx % MAXGT;
  const float* t = target + (size_t)(b * MAXGT + g) * 5;
  float t0 = t[0], t1 = t[1], t2 = t[2], t3 = t[3], t4 = t[4];
  float gx = t1 * NW, gy = t2 * NH, gw = t3 * NW, gh = t4 * NH;
  bool valid = t1 > 0.0f;

  // best anchor by wh-IoU (centered boxes -> inter = min(gw,aw)*min(gh,ah))
  int bn = 0;
  float best = -1.0f, awb = anchors[0], ahb = anchors[1];
  for (int n = 0; n < NA; ++n) {
    float aw = anchors[2 * n], ah = anchors[2 * n + 1];
    float inter = fminf(gw, aw) * fminf(gh, ah);
    float uni = gw * gh + aw * ah - inter;
    float iou = (uni > 0.0f) ? inter / uni : 0.0f;
    if (iou > best) { best = iou; bn = n; awb = aw; ahb = ah; }
  }
  int gi = min(max((int)gx, 0), NW - 1);  // trunc-toward-zero like astype(int32)
  int gj = min(max((int)gy, 0), NH - 1);
  int flat = ((b * NA + bn) * NH + gj) * NW + gi;

  float* p = prep + (size_t)idx * PREP_STRIDE;
  p[0] = gx; p[1] = gy; p[2] = gw; p[3] = gh;
  p[4] = gx - (float)gi;
  p[5] = gy - (float)gj;
  p[6] = __logf(fmaxf(gw, 1e-12f) / awb);
  p[7] = __logf(fmaxf(gh, 1e-12f) / ahb);
  p[8] = valid ? (float)flat : -1.0f;
  p[9] = (float)((int)t0);
  p[10] = 0.0f; p[11] = 0.0f;
}

// ---------------------------------------------------------------------------
// Kernel B: one thread per cell. Coalesced B32 loads of channels 0..4,
// GT metadata staged in LDS, per-block deterministic reduction to partials.
// ---------------------------------------------------------------------------
__global__ void yolo_main(const float* __restrict__ output,
                          const float* __restrict__ anchors,
                          const float* __restrict__ prep,
                          float* __restrict__ partials) {
  __shared__ float sh[MAXGT * PREP_STRIDE];
  __shared__ float wsum[8];

  int gidx = blockIdx.x * 256 + threadIdx.x;   // flat cell index (b,a,j,i)
  int b = gidx / CELLS_PER_B;
  int r = gidx % CELLS_PER_B;
  int a = r / HW;
  int cell = r % HW;
  int j = cell / NW, i = cell % NW;

  // stage this batch's 50 prep entries (600 f32) into LDS
  const float* pb = prep + (size_t)b * MAXGT * PREP_STRIDE;
  for (int k = threadIdx.x; k < MAXGT * PREP_STRIDE; k += 256) sh[k] = pb[k];
  __syncthreads();

  const float* base = output + ((size_t)b * 125 + (size_t)a * 25) * HW + cell;
  float xr = base[0 * HW];
  float yr = base[1 * HW];
  float w  = base[2 * HW];
  float h  = base[3 * HW];
  float cr = base[4 * HW];

  float x = sigm(xr), y = sigm(yr), conf = sigm(cr);
  float aw = anchors[2 * a], ah = anchors[2 * a + 1];
  float px = x + (float)i, py = y + (float)j;
  float pw = __expf(w) * aw, ph = __expf(h) * ah;
  float ax1 = px - pw * 0.5f, ax2 = px + pw * 0.5f;
  float ay1 = py - ph * 0.5f, ay2 = py + ph * 0.5f;
  float parea = pw * ph;

  float maxIoU = 0.0f, mIoU = 0.0f;
  int matched = -1;  // last GT whose scatter cell == this cell (last-writer-wins)
  for (int g = 0; g < MAXGT; ++g) {
    const float* e = sh + g * PREP_STRIDE;
    float fl = e[8];
    if (fl < 0.0f) continue;
    float gx = e[0], gy = e[1], gw = e[2], gh = e[3];
    float bx1 = gx - gw * 0.5f, bx2 = gx + gw * 0.5f;
    float by1 = gy - gh * 0.5f, by2 = gy + gh * 0.5f;
    float iw = fmaxf(fminf(ax2, bx2) - fmaxf(ax1, bx1), 0.0f);
    float ih = fmaxf(fminf(ay2, by2) - fmaxf(ay1, by1), 0.0f);
    float inter = iw * ih;
    float uni = parea + gw * gh - inter;
    float iou = (uni > 0.0f) ? inter / uni : 0.0f;
    maxIoU = fmaxf(maxIoU, iou);
    if ((int)fl == gidx) { matched = g; mIoU = iou; }
  }

  float tx = 0.5f, ty = 0.5f, tw = 0.0f, th = 0.0f, tconf = 0.0f;
  float cmask = (maxIoU > SIL_THRESH_C) ? 0.0f : 1.0f;
  float clsLoss = 0.0f;
  if (matched >= 0) {
    const float* e = sh + matched * PREP_STRIDE;
    tx = e[4]; ty = e[5]; tw = e[6]; th = e[7];
    cmask = OBJECT_SCALE_C;
    tconf = mIoU;
    int tcls = min(max((int)e[9], 0), NC - 1);
    const float* cbase = base + 5 * HW;
    // two passes, no dynamically-indexed register array (avoid scratch)
    float m = -1e30f, tl = 0.0f;
    for (int c = 0; c < NC; ++c) {
      float l = cbase[(size_t)c * HW];
      m = fmaxf(m, l);
      if (c == tcls) tl = l;
    }
    float s = 0.0f;
    for (int c = 0; c < NC; ++c) s += __expf(cbase[(size_t)c * HW] - m);
    clsLoss = (m + __logf(s)) - tl;  // CLASS_SCALE = 1
  }

  float dx = x - tx, dy = y - ty, dw = w - tw, dh = h - th, dc = conf - tconf;
  float loss = 0.5f * (dx * dx + dy * dy + dw * dw + dh * dh)
             + 0.5f * cmask * dc * dc + clsLoss;

  // wave32 reduce, then 8 wave leaders -> LDS -> thread 0 (fixed order, deterministic)
  for (int off = 16; off >= 1; off >>= 1) loss += __shfl_down(loss, off, 32);
  if ((threadIdx.x & 31) == 0) wsum[threadIdx.x >> 5] = loss;
  __syncthreads();
  if (threadIdx.x == 0) {
    float s = 0.0f;
    for (int k = 0; k < 8; ++k) s += wsum[k];
    partials[blockIdx.x] = s;
  }
}

// ---------------------------------------------------------------------------
// Kernel C: deterministic final reduction. One full wave32, all lanes active.
// Ones-matrix V_WMMA_F32_16X16X4_F32: with A = ones(16x4), D[m][n] = sum_k B[k][n],
// so the 32 lane partials (B vgpr0; vgpr1=0) collapse to 16 column sums that
// appear in D vgpr0 lanes 0..15; 4 shuffles finish the sum.
// ---------------------------------------------------------------------------
__global__ void yolo_reduce(const float* __restrict__ partials,
                            float* __restrict__ out, int n) {
  int lane = threadIdx.x;  // 32 threads, EXEC all-ones for the WMMA
  float s = 0.0f;
  for (int k = lane; k < n; k += 32) s += partials[k];

  v2f a; a.x = 1.0f; a.y = 1.0f;   // A = ones (16x4 f32, 2 VGPRs)
  v2f bvec; bvec.x = s; bvec.y = 0.0f;
  v8f c = {};
  v8f d = __builtin_amdgcn_wmma_f32_16x16x4_f32(
      /*neg_a=*/false, a, /*neg_b=*/false, bvec,
      /*c_mod=*/(short)0, c, /*reuse_a=*/false, /*reuse_b=*/false);

  float t = d[0];                   // lane L holds colsum(L % 16) (rows identical)
  t += __shfl_xor(t, 1, 32);
  t += __shfl_xor(t, 2, 32);
  t += __shfl_xor(t, 4, 32);
  t += __shfl_xor(t, 8, 32);        // lane 0 now = sum of the 16 column sums
  if (lane == 0) out[0] = t;
}

extern "C" void kernel_launch(void* const* d_in, const int* in_sizes, int n_in,
                              void* d_out, int out_size, void* d_ws, size_t ws_size,
                              hipStream_t stream) {
  const float* output  = (const float*)d_in[0];
  const float* target  = (const float*)d_in[1];
  const float* anchors = (const float*)d_in[2];

  float* prep     = (float*)d_ws;                       // 800 * 12 f32 = 38.4 KB
  float* partials = prep + (size_t)NB * MAXGT * PREP_STRIDE;  // 1280 f32

  yolo_prep<<<(NB * MAXGT + 255) / 256, 256, 0, stream>>>(target, anchors, prep);
  yolo_main<<<NBLOCKS, 256, 0, stream>>>(output, anchors, prep, partials);
  yolo_reduce<<<1, 32, 0, stream>>>(partials, (float*)d_out, NBLOCKS);
}